// GCNIILayer_87866440941693
// MI455X (gfx1250) — compile-verified
//
#include <hip/hip_runtime.h>

#define N_NODES 100000
#define N_EDGES 1200000
#define DIM 64
#define ALPHA 0.1f
// BETA = 0.5 -> out = relu(0.5*(h3 + h3@W^T))

typedef float v2f __attribute__((ext_vector_type(2)));
typedef float v8f __attribute__((ext_vector_type(8)));

#define LDS_STRIDE 68  // 64 + 4: bank-conflict-free column access, keeps 16B alignment

// ---------------- zero scratch (agg + deg must be zero every call) --------------
__global__ void zero_ws_kernel(float* __restrict__ p, int n) {
    int i = blockIdx.x * blockDim.x + threadIdx.x;
    int stride = gridDim.x * blockDim.x;
    for (; i < n; i += stride) p[i] = 0.0f;
}

// ---------------- in-degree histogram ----------------
__global__ void deg_kernel(const int* __restrict__ dst, float* __restrict__ deg) {
    int e = blockIdx.x * blockDim.x + threadIdx.x;
    if (e < N_EDGES) unsafeAtomicAdd(&deg[dst[e]], 1.0f);  // global_atomic_add_f32
}

// ---------------- norm = clamp(deg,1)^-0.5 ----------------
__global__ void norm_kernel(const float* __restrict__ deg, float* __restrict__ norm) {
    int n = blockIdx.x * blockDim.x + threadIdx.x;
    if (n < N_NODES) {
        float d = deg[n];
        d = d < 1.0f ? 1.0f : d;
        norm[n] = 1.0f / sqrtf(d);
    }
}

// ---------------- hscaled = features * norm[row] (float4) ----------------
__global__ void scale_kernel(const float* __restrict__ feat,
                             const float* __restrict__ norm,
                             float* __restrict__ hs) {
    int i = blockIdx.x * blockDim.x + threadIdx.x;  // over N*D/4 float4s
    if (i < N_NODES * (DIM / 4)) {
        int node = i / (DIM / 4);
        float nv = norm[node];
        float4 f = ((const float4*)feat)[i];
        float4 o;
        o.x = f.x * nv; o.y = f.y * nv; o.z = f.z * nv; o.w = f.w * nv;
        ((float4*)hs)[i] = o;
    }
}

// ---------------- edge scatter: agg[dst] += hscaled[src]  (16 threads / edge) ----
__global__ void agg_kernel(const int* __restrict__ src, const int* __restrict__ dst,
                           const float* __restrict__ hs, float* __restrict__ agg) {
    long long t = (long long)blockIdx.x * blockDim.x + threadIdx.x;
    int e = (int)(t >> 4);
    if (e >= N_EDGES) return;
    int c = ((int)t & 15) * 4;
    int s = src[e];
    int d = dst[e];
    float4 v = *(const float4*)(hs + (size_t)s * DIM + c);
    float* ap = agg + (size_t)d * DIM + c;
    unsafeAtomicAdd(ap + 0, v.x);
    unsafeAtomicAdd(ap + 1, v.y);
    unsafeAtomicAdd(ap + 2, v.z);
    unsafeAtomicAdd(ap + 3, v.w);
}

// ---------------- fused epilogue: norm-scale + residual + (h3 @ W^T) + ReLU ------
// Block = 128 threads = 4 wave32s, handles 16 nodes. Wave w owns output cols
// [16w, 16w+16). Per-wave K loop: 16x V_WMMA_F32_16X16X4_F32 over K=64.
__global__ __launch_bounds__(128) void gcnii_final_kernel(
        const float* __restrict__ agg, const float* __restrict__ norm,
        const float* __restrict__ initf, const float* __restrict__ W,
        float* __restrict__ out) {
    __shared__ float sH[16 * LDS_STRIDE];   // h3 tile: 16 nodes x 64
    __shared__ float sW[64 * LDS_STRIDE];   // full W: 64 x 64 (row = out, col = in)

    const int tid = threadIdx.x;
    const int nodeBase = blockIdx.x * 16;   // N == 6250*16, no partial tiles

    // Stage W into LDS: 1024 float4s, 8 per thread
    for (int i = tid; i < 64 * 16; i += 128) {
        int r = i >> 4;
        int c4 = i & 15;
        float4 w = ((const float4*)W)[i];
        float* p = &sW[r * LDS_STRIDE + c4 * 4];
        p[0] = w.x; p[1] = w.y; p[2] = w.z; p[3] = w.w;
    }
    // Build h3 tile: h3 = 0.9*(agg*norm) + 0.1*init ; 256 float4s, 2 per thread
    for (int i = tid; i < 16 * 16; i += 128) {
        int r = i >> 4;          // local node
        int c4 = i & 15;         // float4 column
        int node = nodeBase + r;
        float nv = norm[node];
        float4 a = ((const float4*)(agg  + (size_t)node * DIM))[c4];
        float4 f = ((const float4*)(initf + (size_t)node * DIM))[c4];
        float* p = &sH[r * LDS_STRIDE + c4 * 4];
        p[0] = (1.0f - ALPHA) * (a.x * nv) + ALPHA * f.x;
        p[1] = (1.0f - ALPHA) * (a.y * nv) + ALPHA * f.y;
        p[2] = (1.0f - ALPHA) * (a.z * nv) + ALPHA * f.z;
        p[3] = (1.0f - ALPHA) * (a.w * nv) + ALPHA * f.w;
    }
    __syncthreads();

    const int lane  = tid & 31;
    const int wave  = tid >> 5;
    const int j0    = wave * 16;        // output-column tile base
    const int nIdx  = lane & 15;        // N index (cols) / M index for A rows
    const int koff  = (lane >> 4) * 2;  // A/B hold K pair {koff, koff+1} per VGPR pair
    const int mbase = (lane >> 4) * 8;  // C/D: lanes 16-31 hold M = v+8

    // Seed accumulator with h3[M][j0+N] so result = 0.5*(h3 + h3@W^T)
    v8f acc;
#pragma unroll
    for (int v = 0; v < 8; ++v)
        acc[v] = sH[(v + mbase) * LDS_STRIDE + (j0 + nIdx)];

#pragma unroll
    for (int k0 = 0; k0 < DIM; k0 += 4) {
        v2f a, b;
        // A (16x4): lane<16 -> M=lane, K={k0,k0+1}; lane>=16 -> M=lane-16, K={k0+2,k0+3}
        a.x = sH[nIdx * LDS_STRIDE + k0 + koff];
        a.y = sH[nIdx * LDS_STRIDE + k0 + koff + 1];
        // B (4x16): B[k][n] = W[j0+n][k] (h3 @ W^T)
        b.x = sW[(j0 + nIdx) * LDS_STRIDE + k0 + koff];
        b.y = sW[(j0 + nIdx) * LDS_STRIDE + k0 + koff + 1];
        acc = __builtin_amdgcn_wmma_f32_16x16x4_f32(
            /*neg_a=*/false, a, /*neg_b=*/false, b,
            /*c_mod=*/(short)0, acc, /*reuse_a=*/false, /*reuse_b=*/false);
    }

#pragma unroll
    for (int v = 0; v < 8; ++v) {
        int node = nodeBase + v + mbase;
        float r = 0.5f * acc[v];
        out[(size_t)node * DIM + j0 + nIdx] = r > 0.0f ? r : 0.0f;
    }
}

extern "C" void kernel_launch(void* const* d_in, const int* in_sizes, int n_in,
                              void* d_out, int out_size, void* d_ws, size_t ws_size,
                              hipStream_t stream) {
    const float* features = (const float*)d_in[0];
    const float* initf    = (const float*)d_in[1];
    const float* W        = (const float*)d_in[2];
    const int*   src      = (const int*)d_in[3];
    const int*   dst      = (const int*)d_in[4];
    float* out = (float*)d_out;

    // Workspace layout (floats): [agg N*D][deg N][hs N*D][norm N]
    float* agg  = (float*)d_ws;
    float* deg  = agg + (size_t)N_NODES * DIM;
    float* hs   = deg + N_NODES;
    float* norm = hs + (size_t)N_NODES * DIM;

    // 1) zero agg + deg (contiguous)
    zero_ws_kernel<<<2048, 256, 0, stream>>>(agg, N_NODES * DIM + N_NODES);
    // 2) in-degree
    deg_kernel<<<(N_EDGES + 255) / 256, 256, 0, stream>>>(dst, deg);
    // 3) norm
    norm_kernel<<<(N_NODES + 255) / 256, 256, 0, stream>>>(deg, norm);
    // 4) row scale
    scale_kernel<<<(N_NODES * (DIM / 4) + 255) / 256, 256, 0, stream>>>(features, norm, hs);
    // 5) edge scatter-add (16 threads per edge)
    agg_kernel<<<(int)(((long long)N_EDGES * 16 + 255) / 256), 256, 0, stream>>>(src, dst, hs, agg);
    // 6) fused WMMA epilogue
    gcnii_final_kernel<<<N_NODES / 16, 128, 0, stream>>>(agg, norm, initf, W, out);
}